// GNNModel_35278861369969
// MI455X (gfx1250) — compile-verified
//
#include <hip/hip_runtime.h>

#define D_DIM 64
#define A_DIM 5

typedef __attribute__((ext_vector_type(2))) float v2f;
typedef __attribute__((ext_vector_type(8))) float v8f;

// ---------------------------------------------------------------- zero agg
__global__ void zero_f32(float* __restrict__ p, int n) {
  int i = blockIdx.x * blockDim.x + threadIdx.x;
  if (i < n) p[i] = 0.0f;
}

// --------------------------------------------- out[i][a] = X[i,:] . W[a,:]
// X: [n,64], W: [5,64], out: [n,5]
__global__ void proj5_kernel(const float* __restrict__ X, const float* __restrict__ W,
                             float* __restrict__ out, int n) {
  int i = blockIdx.x * blockDim.x + threadIdx.x;
  if (i >= n) return;
  const float4* x = (const float4*)(X + (size_t)i * D_DIM);
  float acc[A_DIM] = {0.f, 0.f, 0.f, 0.f, 0.f};
  #pragma unroll 4
  for (int k = 0; k < D_DIM / 4; ++k) {
    float4 v = x[k];
    #pragma unroll
    for (int a = 0; a < A_DIM; ++a) {
      float4 w = ((const float4*)(W + a * D_DIM))[k];
      acc[a] += v.x * w.x + v.y * w.y + v.z * w.z + v.w * w.w;
    }
  }
  #pragma unroll
  for (int a = 0; a < A_DIM; ++a) out[(size_t)i * A_DIM + a] = acc[a];
}

// ------------- proj_q[b][a] = rela_embed[q_rel[b],:] . Wqr_w[a,:] + Wqr_b[a]
__global__ void projq_kernel(const float* __restrict__ rela, const int* __restrict__ q_rel,
                             const float* __restrict__ W, const float* __restrict__ bias,
                             float* __restrict__ out, int Bn) {
  int b = blockIdx.x * blockDim.x + threadIdx.x;
  if (b >= Bn) return;
  const float4* x = (const float4*)(rela + (size_t)q_rel[b] * D_DIM);
  float acc[A_DIM];
  #pragma unroll
  for (int a = 0; a < A_DIM; ++a) acc[a] = bias[a];
  #pragma unroll 4
  for (int k = 0; k < D_DIM / 4; ++k) {
    float4 v = x[k];
    #pragma unroll
    for (int a = 0; a < A_DIM; ++a) {
      float4 w = ((const float4*)(W + a * D_DIM))[k];
      acc[a] += v.x * w.x + v.y * w.y + v.z * w.z + v.w * w.w;
    }
  }
  #pragma unroll
  for (int a = 0; a < A_DIM; ++a) out[(size_t)b * A_DIM + a] = acc[a];
}

// --------------------------- alpha[e] = sigmoid(wa . relu(ps+pr+pq) + wa_b)
__global__ void alpha_kernel(const float* __restrict__ proj_s, const float* __restrict__ proj_r,
                             const float* __restrict__ proj_q,
                             const float* __restrict__ wa_w, const float* __restrict__ wa_b,
                             const int* __restrict__ sub, const int* __restrict__ rel,
                             const int* __restrict__ r_idx,
                             float* __restrict__ alpha, int E) {
  int e = blockIdx.x * blockDim.x + threadIdx.x;
  if (e >= E) return;
  int s = sub[e], r = rel[e], q = r_idx[e];
  float z = wa_b[0];
  #pragma unroll
  for (int a = 0; a < A_DIM; ++a) {
    float p = proj_s[(size_t)s * A_DIM + a] + proj_r[(size_t)r * A_DIM + a]
            + proj_q[(size_t)q * A_DIM + a];
    z = fmaf(fmaxf(p, 0.0f), wa_w[a], z);
  }
  alpha[e] = 1.0f / (1.0f + __expf(-z));
}

// -------------- agg[obj[e],:] += alpha[e]*(hs+hr)   (16 threads per edge)
__global__ void scatter_kernel(const float* __restrict__ layer_input,
                               const float* __restrict__ rela_embed,
                               const float* __restrict__ alpha,
                               const int* __restrict__ sub, const int* __restrict__ rel,
                               const int* __restrict__ obj,
                               float* __restrict__ agg, int E) {
  int gid = blockIdx.x * blockDim.x + threadIdx.x;
  int e = gid >> 4;
  if (e >= E) return;
  int t = gid & 15;
  float al = alpha[e];
  int s = sub[e], r = rel[e], o = obj[e];
  float4 hs = ((const float4*)(layer_input + (size_t)s * D_DIM))[t];
  float4 hr = ((const float4*)(rela_embed + (size_t)r * D_DIM))[t];
  float* dst = agg + (size_t)o * D_DIM + t * 4;
  atomicAdd(dst + 0, al * (hs.x + hr.x));
  atomicAdd(dst + 1, al * (hs.y + hr.y));
  atomicAdd(dst + 2, al * (hs.z + hr.z));
  atomicAdd(dst + 3, al * (hs.w + hr.w));
}

// ------------------------- Out = relu(Agg @ Wh.T), WMMA f32 16x16x4, K=64
// One wave per 16x16 output tile. A = Agg tile (M x K), B[k][n] = Wh[n][k].
__global__ void gemm_relu_wmma(const float* __restrict__ Agg, const float* __restrict__ Wh,
                               float* __restrict__ Out, int Nrows) {
  int wave = (int)((blockIdx.x * (unsigned)blockDim.x + threadIdx.x) >> 5);
  int lane = threadIdx.x & 31;
  int rowTile = wave >> 2;      // 64/16 = 4 column tiles
  int colTile = wave & 3;
  int row0 = rowTile * 16;
  if (row0 >= Nrows) return;    // wave-uniform; surviving waves keep EXEC all-1s
  int col0 = colTile * 16;
  int lhalf = lane >> 4;        // 0: K pair {0,1}; 1: K pair {2,3}
  int lmod  = lane & 15;

  int arow = row0 + lmod;
  if (arow >= Nrows) arow = Nrows - 1;          // clamp keeps EXEC full
  const float* aptr = Agg + (size_t)arow * D_DIM + lhalf * 2;
  const float* bptr = Wh  + (size_t)(col0 + lmod) * D_DIM + lhalf * 2;

  v8f c = {};
  #pragma unroll
  for (int k0 = 0; k0 < D_DIM; k0 += 4) {
    v2f a, b;
    a.x = aptr[k0 + 0]; a.y = aptr[k0 + 1];
    b.x = bptr[k0 + 0]; b.y = bptr[k0 + 1];
    c = __builtin_amdgcn_wmma_f32_16x16x4_f32(
        /*neg_a=*/false, a, /*neg_b=*/false, b,
        /*c_mod=*/(short)0, c, /*reuse_a=*/false, /*reuse_b=*/false);
  }

  #pragma unroll
  for (int r = 0; r < 8; ++r) {
    int row = row0 + r + lhalf * 8;
    if (row < Nrows) {
      float v = c[r];
      Out[(size_t)row * D_DIM + col0 + lmod] = v > 0.0f ? v : 0.0f;
    }
  }
}

// ---------------------------------------------------------------- launch
extern "C" void kernel_launch(void* const* d_in, const int* in_sizes, int n_in,
                              void* d_out, int out_size, void* d_ws, size_t ws_size,
                              hipStream_t stream) {
  const float* layer_input = (const float*)d_in[0];
  const float* rela_embed  = (const float*)d_in[1];
  const float* Ws    = (const float*)d_in[2];
  const float* Wr    = (const float*)d_in[3];
  const float* Wqr_w = (const float*)d_in[4];
  const float* Wqr_b = (const float*)d_in[5];
  const float* wa_w  = (const float*)d_in[6];
  const float* wa_b  = (const float*)d_in[7];
  const float* Wh    = (const float*)d_in[8];
  const int* sub   = (const int*)d_in[9];
  const int* rel   = (const int*)d_in[10];
  const int* obj   = (const int*)d_in[11];
  const int* r_idx = (const int*)d_in[12];
  const int* q_rel = (const int*)d_in[13];

  const int N   = in_sizes[0] / D_DIM;
  const int NR2 = in_sizes[1] / D_DIM;
  const int E   = in_sizes[9];
  const int Bn  = in_sizes[13];

  // workspace layout (floats): agg[N*64] | proj_s[N*5] | proj_r[NR2*5] | proj_q[B*5] | alpha[E]
  float* ws     = (float*)d_ws;
  float* agg    = ws;
  float* proj_s = agg    + (size_t)N * D_DIM;
  float* proj_r = proj_s + (size_t)N * A_DIM;
  float* proj_q = proj_r + (size_t)NR2 * A_DIM;
  float* alpha  = proj_q + (size_t)Bn * A_DIM;

  const int aggN = N * D_DIM;
  zero_f32<<<(aggN + 255) / 256, 256, 0, stream>>>(agg, aggN);

  proj5_kernel<<<(N + 127) / 128, 128, 0, stream>>>(layer_input, Ws, proj_s, N);
  proj5_kernel<<<(NR2 + 127) / 128, 128, 0, stream>>>(rela_embed, Wr, proj_r, NR2);
  projq_kernel<<<(Bn + 127) / 128, 128, 0, stream>>>(rela_embed, q_rel, Wqr_w, Wqr_b, proj_q, Bn);

  alpha_kernel<<<(E + 255) / 256, 256, 0, stream>>>(proj_s, proj_r, proj_q, wa_w, wa_b,
                                                    sub, rel, r_idx, alpha, E);

  {
    long long threads = (long long)E * 16;
    int blocks = (int)((threads + 255) / 256);
    scatter_kernel<<<blocks, 256, 0, stream>>>(layer_input, rela_embed, alpha,
                                               sub, rel, obj, agg, E);
  }

  {
    int rowTiles = (N + 15) / 16;
    long long waves = (long long)rowTiles * 4;
    int blocks = (int)((waves + 7) / 8);     // 8 wave32s per 256-thread block
    gemm_relu_wmma<<<blocks, 256, 0, stream>>>(agg, Wh, (float*)d_out, N);
  }
}